// FocalLoss_77489799954599
// MI455X (gfx1250) — compile-verified
//
#include <hip/hip_runtime.h>

typedef float f32x4 __attribute__((ext_vector_type(4)));
typedef int   i32x4 __attribute__((ext_vector_type(4)));
typedef float f32x2 __attribute__((ext_vector_type(2)));
typedef float f32x8 __attribute__((ext_vector_type(8)));

#define NPART 4096
#define BLK   256

// ---------------------------------------------------------------------------
// Kernel 1: streaming focal-loss partial reduction.
//   per element: t==1 -> log(x)*(1-x)^2*0.75 ; t==0 -> log(1-x)*x^2*0.25
//   (gamma == 2.0 -> square, no powf; one v_log_f32 per element via argument
//   selection). Non-temporal b128 loads: 268 MB stream > 192 MB L2.
// ---------------------------------------------------------------------------
__global__ __launch_bounds__(BLK) void focal_partial(const float* __restrict__ in,
                                                     const int*   __restrict__ tg,
                                                     float* __restrict__ ploss,
                                                     int*   __restrict__ pcnt,
                                                     int n)
{
    const int tid    = threadIdx.x;
    const int gid    = blockIdx.x * BLK + tid;
    const int stride = gridDim.x * BLK;
    const int nvec   = n >> 2;

    const f32x4* in4 = (const f32x4*)in;
    const i32x4* tg4 = (const i32x4*)tg;

    float loss = 0.0f;
    int   cnt  = 0;

    for (int i = gid; i < nvec; i += stride) {
        f32x4 x = __builtin_nontemporal_load(&in4[i]);
        i32x4 t = __builtin_nontemporal_load(&tg4[i]);
#pragma unroll
        for (int j = 0; j < 4; ++j) {
            float xv  = x[j];
            int   tv  = t[j];
            bool  pos = (tv == 1);
            float a   = pos ? xv : (1.0f - xv);   // log argument
            float m   = 1.0f - a;                 // pos: (1-x), neg: x
            float w   = pos ? 0.75f : 0.25f;
            loss += __logf(a) * (m * m) * w;
            // jnp.round == round-half-to-even -> v_rndne_f32
            cnt  += (rintf(xv) == (float)tv) ? 1 : 0;
        }
    }
    // scalar tail (N is a power of two, but stay safe)
    for (int i = (nvec << 2) + gid; i < n; i += stride) {
        float xv  = in[i];
        int   tv  = tg[i];
        bool  pos = (tv == 1);
        float a   = pos ? xv : (1.0f - xv);
        float m   = 1.0f - a;
        float w   = pos ? 0.75f : 0.25f;
        loss += __logf(a) * (m * m) * w;
        cnt  += (rintf(xv) == (float)tv) ? 1 : 0;
    }

    // wave32 butterfly reduction
#pragma unroll
    for (int off = 16; off > 0; off >>= 1) {
        loss += __shfl_down(loss, off, 32);
        cnt  += __shfl_down(cnt,  off, 32);
    }

    __shared__ float wl[BLK / 32];
    __shared__ int   wc[BLK / 32];
    const int wave = tid >> 5;
    const int lane = tid & 31;
    if (lane == 0) { wl[wave] = loss; wc[wave] = cnt; }
    __syncthreads();
    if (tid == 0) {
        float l = 0.0f; int c = 0;
#pragma unroll
        for (int wv = 0; wv < BLK / 32; ++wv) { l += wl[wv]; c += wc[wv]; }
        ploss[blockIdx.x] = l;
        pcnt[blockIdx.x]  = c;
    }
}

// ---------------------------------------------------------------------------
// Kernel 2: deterministic final reduction of NPART partials.
//   Loss path uses V_WMMA_F32_16X16X4_F32 on wave 0: D = ones(16x4) * B + C
//   sums 64 values per issue into per-column sums; A==1 everywhere makes the
//   B-operand VGPR striping irrelevant to the total. Summing D's VGPR0 over
//   all 32 lanes double-counts (rows M=0 and M=8) -> scale by 0.5.
// ---------------------------------------------------------------------------
__global__ __launch_bounds__(BLK) void focal_final(const float* __restrict__ ploss,
                                                   const int*   __restrict__ pcnt,
                                                   float* __restrict__ out,
                                                   int n)
{
    __shared__ float sl[BLK];
    __shared__ int   sc[BLK];
    const int tid = threadIdx.x;

    float l = 0.0f; int c = 0;
#pragma unroll
    for (int k = 0; k < NPART / BLK; ++k) {
        l += ploss[tid + k * BLK];
        c += pcnt[tid + k * BLK];
    }
    sl[tid] = l;
    sc[tid] = c;
    __syncthreads();

    // exact integer count: LDS tree (deterministic)
    for (int s = BLK / 2; s > 0; s >>= 1) {
        if (tid < s) sc[tid] += sc[tid + s];
        __syncthreads();
    }

    if (tid < 32) {   // wave 0 only: EXEC all-ones inside this wave
        f32x2 a;  a.x = 1.0f; a.y = 1.0f;   // A = ones(16x4)
        f32x8 acc = {};
#pragma unroll
        for (int ch = 0; ch < 4; ++ch) {
            f32x2 b;
            b.x = sl[ch * 64 + tid];
            b.y = sl[ch * 64 + 32 + tid];
            acc = __builtin_amdgcn_wmma_f32_16x16x4_f32(
                false, a, false, b, (short)0, acc, false, false);
        }
        float v = acc[0];
#pragma unroll
        for (int off = 16; off > 0; off >>= 1) v += __shfl_xor(v, off, 32);
        v *= 0.5f;   // rows M=0 and M=8 both present in VGPR0 -> halve

        if (tid == 0) {
            out[0] = -v / (float)n;               // loss
            out[1] = (float)sc[0] / (float)n;     // accuracy
        }
    }
}

extern "C" void kernel_launch(void* const* d_in, const int* in_sizes, int n_in,
                              void* d_out, int out_size, void* d_ws, size_t ws_size,
                              hipStream_t stream) {
    const float* in = (const float*)d_in[0];
    const int*   tg = (const int*)d_in[1];
    const int n = in_sizes[0];

    float* ploss = (float*)d_ws;
    int*   pcnt  = (int*)((char*)d_ws + NPART * sizeof(float));

    focal_partial<<<NPART, BLK, 0, stream>>>(in, tg, ploss, pcnt, n);
    focal_final<<<1, BLK, 0, stream>>>(ploss, pcnt, (float*)d_out, n);
}